// KNNLayer_80934363726401
// MI455X (gfx1250) — compile-verified
//
#include <hip/hip_runtime.h>

typedef __attribute__((ext_vector_type(2))) float v2f;
typedef __attribute__((ext_vector_type(4))) float v4f;
typedef __attribute__((ext_vector_type(8))) float v8f;

#define N_PTS   8192
#define BATCH   4
#define NK      16
#define THREADS 256
#define QPB     128   // queries per block: 8 waves * 16

__global__ __launch_bounds__(THREADS) void knn_wmma_kernel(
    const float* __restrict__ pts_g,   // [B, N, 3]
    int* __restrict__ out)             // [B, N, K]
{
    extern __shared__ char smem[];
    float4* pts = reinterpret_cast<float4*>(smem);                       // 128 KB: (x,y,z,|p|^2)
    float*  sqc = reinterpret_cast<float*>(smem + N_PTS * sizeof(float4)); // 32 KB: |p|^2

    const int b = blockIdx.y;
    const float* pb = pts_g + (size_t)b * N_PTS * 3;

    // ---- cooperative load of the whole batch into LDS ----
    for (int i = threadIdx.x; i < N_PTS; i += THREADS) {
        float x = pb[i * 3 + 0];
        float y = pb[i * 3 + 1];
        float z = pb[i * 3 + 2];
        float s = x * x + y * y + z * z;
        pts[i] = make_float4(x, y, z, s);
        sqc[i] = s;
    }
    __syncthreads();

    const int lane  = threadIdx.x & 31;
    const int wave  = threadIdx.x >> 5;
    const int qbase = blockIdx.x * QPB + wave * 16;
    const int q     = qbase + (lane & 15);
    const bool hi   = lane >= 16;

    // ---- B operand: -2 * query coords (loop-invariant) ----
    // f32 4x16 B layout: lanes 0-15 -> K=0 (v0), K=1 (v1); lanes 16-31 -> K=2 (v0), K=3 (v1)
    float4 qp = pts[q];
    v2f bmat;
    bmat.x = -2.0f * (hi ? qp.z : qp.x);
    bmat.y = hi ? 0.0f : (-2.0f * qp.y);
    const float sqq = qp.w;

    // running sorted top-16 (ascending distance) per lane
    float dist[NK];
    int   idx[NK];
#pragma unroll
    for (int k = 0; k < NK; ++k) { dist[k] = 3.4e38f; idx[k] = 0; }

    const float* sqhalf = sqc + (hi ? 8 : 0);
    const int    moff   = hi ? 8 : 0;

    // ---- main loop: one WMMA per 16-candidate tile ----
    for (int t = 0; t < N_PTS; t += 16) {
        // A operand: candidate coords. Lane L & L+16 read the same float4 (broadcast).
        float4 cp = pts[t + (lane & 15)];
        v2f amat;
        amat.x = hi ? cp.z : cp.x;
        amat.y = hi ? 0.0f : cp.y;

        // C bias: |cand_M|^2 + |query_N|^2 in the 16x16 f32 C/D layout
        const v4f* sp = reinterpret_cast<const v4f*>(sqhalf + t);
        v4f s0 = sp[0];
        v4f s1 = sp[1];
        v8f c;
        c[0] = s0.x + sqq; c[1] = s0.y + sqq; c[2] = s0.z + sqq; c[3] = s0.w + sqq;
        c[4] = s1.x + sqq; c[5] = s1.y + sqq; c[6] = s1.z + sqq; c[7] = s1.w + sqq;

        // D = cand . (-2*query) + (|c|^2 + |q|^2)  == squared distance tile
        v8f d = __builtin_amdgcn_wmma_f32_16x16x4_f32(
            false, amat, false, bmat, (short)0, c, false, false);

        const int ibase = t + moff;
#pragma unroll
        for (int r = 0; r < 8; ++r) {
            float dv = d[r];
            if (dv < dist[NK - 1]) {
                dist[NK - 1] = dv;
                idx[NK - 1]  = ibase + r;
#pragma unroll
                for (int k = NK - 1; k > 0; --k) {
                    if (dist[k] < dist[k - 1]) {
                        float td = dist[k]; dist[k] = dist[k - 1]; dist[k - 1] = td;
                        int   ti = idx[k];  idx[k]  = idx[k - 1];  idx[k - 1]  = ti;
                    }
                }
            }
        }
    }

    // ---- merge lane L with lane L^16 (both hold sorted 16-lists for the same query) ----
    // bitonic partition: min(A[i], B[15-i]) = 16 smallest of the 32, as a bitonic sequence
    float nd[NK];
    int   ni[NK];
#pragma unroll
    for (int i = 0; i < NK; ++i) {
        float od = __shfl_xor(dist[NK - 1 - i], 16, 32);
        int   oi = __shfl_xor(idx[NK - 1 - i], 16, 32);
        if (od < dist[i]) { nd[i] = od;      ni[i] = oi;     }
        else              { nd[i] = dist[i]; ni[i] = idx[i]; }
    }
    // bitonic merge network (static indices after unroll)
#pragma unroll
    for (int s = 8; s >= 1; s >>= 1) {
#pragma unroll
        for (int i = 0; i < NK; ++i) {
            if ((i & s) == 0) {
                int j = i | s;
                if (nd[j] < nd[i]) {
                    float td = nd[i]; nd[i] = nd[j]; nd[j] = td;
                    int   ti = ni[i]; ni[i] = ni[j]; ni[j] = ti;
                }
            }
        }
    }

    // lanes 0-15 write the K indices (ascending distance, matching top_k(-adj) order)
    if (lane < 16) {
        int* o = out + ((size_t)b * N_PTS + q) * NK;
#pragma unroll
        for (int i = 0; i < NK; ++i) o[i] = ni[i];
    }
}

extern "C" void kernel_launch(void* const* d_in, const int* in_sizes, int n_in,
                              void* d_out, int out_size, void* d_ws, size_t ws_size,
                              hipStream_t stream) {
    const float* pts = (const float*)d_in[0];
    int* out = (int*)d_out;

    dim3 grid(N_PTS / QPB, BATCH);
    dim3 block(THREADS);
    size_t shmem = (size_t)N_PTS * sizeof(float4) + (size_t)N_PTS * sizeof(float); // 160 KB
    knn_wmma_kernel<<<grid, block, shmem, stream>>>(pts, out);
}